// TrajPredictor_81183471829406
// MI455X (gfx1250) — compile-verified
//
#include <hip/hip_runtime.h>
#include <cmath>

// ---------------------------------------------------------------------------
// TrajPredictor RNN (encoder 50 steps + autoregressive decoder 12 steps)
// fp32 WMMA 16x16x4 on gfx1250, one 16-row tile per wave, W_hh streamed
// from bank-conflict-free padded LDS, tanh via v_tanh_f32.
// ---------------------------------------------------------------------------

typedef __attribute__((ext_vector_type(2))) float v2f;
typedef __attribute__((ext_vector_type(8))) float v8f;

#define I_SIZE   3
#define H        128
#define O_SIZE   3
#define BATCH    512
#define T_STEPS  50
#define AGENTS   64
#define PRED     12
#define NROWS    (BATCH * AGENTS)          // 32768 independent RNN rows
#define WPAD     132                       // row padding: 132 mod 64 = 4 -> conflict-free b64
#define WAVES    8
#define TILE_M   16
#define ROWS_PER_BLOCK (WAVES * TILE_M)    // 128
#define NBLOCKS  (NROWS / ROWS_PER_BLOCK)  // 256

struct __align__(16) Smem {
    float W[H][WPAD];                  // current W_hh (encoder phase, then decoder phase)
    float fcT[H][16];                  // fcT[k][n] = fc_W[n][k], cols 3..15 zero
    float stage[WAVES][TILE_M][WPAD];  // per-wave C-layout -> A-layout transpose buffer
};
// 67584 + 8192 + 67584 = 143360 bytes

__device__ __forceinline__ v2f mk2(float a, float b) { v2f r; r[0] = a; r[1] = b; return r; }

__device__ __forceinline__ v8f splat8(float x) {
    v8f r;
#pragma unroll
    for (int i = 0; i < 8; ++i) r[i] = x;
    return r;
}

__device__ __forceinline__ v8f wmma4(v2f a, v2f b, v8f c) {
    // D = A(16x4 f32) * B(4x16 f32) + C(16x16 f32)
    return __builtin_amdgcn_wmma_f32_16x16x4_f32(
        /*neg_a=*/false, a, /*neg_b=*/false, b,
        /*c_mod=*/(short)0, c, /*reuse_a=*/false, /*reuse_b=*/false);
}

__device__ __forceinline__ float fast_tanh(float x) {
#if __has_builtin(__builtin_amdgcn_tanhf)
    return __builtin_amdgcn_tanhf(x);   // V_TANH_F32 (CDNA5 TRANS op)
#else
    return ::tanhf(x);
#endif
}

__global__ void __launch_bounds__(256, 1)
traj_rnn_kernel(const float* __restrict__ src,
                const float* __restrict__ W_ih_enc,
                const float* __restrict__ W_hh_enc,
                const float* __restrict__ b_ih_enc,
                const float* __restrict__ b_hh_enc,
                const float* __restrict__ W_ih_dec,
                const float* __restrict__ W_hh_dec,
                const float* __restrict__ b_ih_dec,
                const float* __restrict__ b_hh_dec,
                const float* __restrict__ fc_W,
                const float* __restrict__ fc_b,
                float* __restrict__ out)
{
    __shared__ Smem s;

    const int tid  = threadIdx.x;
    const int wave = tid >> 5;
    const int lane = tid & 31;
    const int l16  = lane & 15;
    const bool hi  = lane >= 16;      // lanes 16..31 hold K+2/K+3 (A) and rows M+8 (C/D)
    const int koff = hi ? 2 : 0;
    const int r0   = blockIdx.x * ROWS_PER_BLOCK + wave * TILE_M;

    // ---- cooperative LDS fill: W_hh_enc row-major (padded), fc_W^T padded to 16 cols
    for (int i = tid; i < H * H; i += 256)
        s.W[i >> 7][i & 127] = W_hh_enc[i];
    for (int i = tid; i < H * 16; i += 256) {
        const int k = i >> 4, n = i & 15;
        s.fcT[k][n] = (n < O_SIZE) ? fc_W[n * H + k] : 0.0f;
    }
    __syncthreads();

    // ---- register-resident B fragments for the K=4-padded x @ W_ih^T term.
    // B[k][n] = W_ih[n][k] (k<3), zero-padded. V0={K0|K2}, V1={K1|K3}.
    v2f BihE[8], BihD[8];
    float biasE[8], biasD[8];
#pragma unroll
    for (int t = 0; t < 8; ++t) {
        const int n = t * 16 + l16;
        if (!hi) {
            BihE[t] = mk2(W_ih_enc[n * I_SIZE + 0], W_ih_enc[n * I_SIZE + 1]);
            BihD[t] = mk2(W_ih_dec[n * I_SIZE + 0], W_ih_dec[n * I_SIZE + 1]);
        } else {
            BihE[t] = mk2(W_ih_enc[n * I_SIZE + 2], 0.0f);
            BihD[t] = mk2(W_ih_dec[n * I_SIZE + 2], 0.0f);
        }
        biasE[t] = b_ih_enc[n] + b_hh_enc[n];
        biasD[t] = b_ih_dec[n] + b_hh_dec[n];
    }
    const float fcb = (l16 < O_SIZE) ? fc_b[l16] : 0.0f;

    // ---- per-lane source row for A fragments (row r = r0 + l16)
    const int myrow = r0 + l16;
    const int mb = myrow >> 6;                 // batch index
    const int ma = myrow & 63;                 // agent index
    const float* xbase = src + (size_t)mb * T_STEPS * (AGENTS * I_SIZE) + ma * I_SIZE;

    // ---- hidden state as 32 A-fragments (16x4 f32 each): h starts at zero
    v2f hA[32];
#pragma unroll
    for (int c = 0; c < 32; ++c) hA[c] = mk2(0.0f, 0.0f);

    v2f Ax = mk2(0.0f, 0.0f);
    v8f acc[8];

    // =========================== encoder: 50 steps ===========================
    for (int t = 0; t < T_STEPS; ++t) {
        const float* xr = xbase + (size_t)t * (AGENTS * I_SIZE);
        Ax = hi ? mk2(xr[2], 0.0f) : mk2(xr[0], xr[1]);

#pragma unroll
        for (int n = 0; n < 8; ++n) acc[n] = splat8(biasE[n]);
#pragma unroll
        for (int n = 0; n < 8; ++n) acc[n] = wmma4(Ax, BihE[n], acc[n]);

        // K-chain: 32 chunks x 8 N-tiles, B streamed from padded LDS (ds_load_b64)
#pragma unroll
        for (int c = 0; c < 32; ++c) {
#pragma unroll
            for (int n = 0; n < 8; ++n) {
                const v2f Bv = *(const v2f*)&s.W[n * 16 + l16][4 * c + koff];
                acc[n] = wmma4(hA[c], Bv, acc[n]);
            }
        }

        // tanh, then C-layout -> A-layout transpose through wave-private LDS.
        // Same-wave DS ops are processed in order: no barrier needed.
#pragma unroll
        for (int n = 0; n < 8; ++n) {
#pragma unroll
            for (int v = 0; v < 8; ++v) {
                const int row = hi ? v + 8 : v;
                s.stage[wave][row][n * 16 + l16] = fast_tanh(acc[n][v]);
            }
        }
#pragma unroll
        for (int c = 0; c < 32; ++c)
            hA[c] = *(const v2f*)&s.stage[wave][l16][4 * c + koff];
    }

    // decoder initial input = last observed x (t = T-1)
    {
        const float* xr = xbase + (size_t)(T_STEPS - 1) * (AGENTS * I_SIZE);
        Ax = hi ? mk2(xr[2], 0.0f) : mk2(xr[0], xr[1]);
    }

    // swap LDS weight buffer to W_hh_dec
    __syncthreads();
    for (int i = tid; i < H * H; i += 256)
        s.W[i >> 7][i & 127] = W_hh_dec[i];
    __syncthreads();

    // ==================== decoder: 12 autoregressive steps ====================
    for (int p = 0; p < PRED; ++p) {
#pragma unroll
        for (int n = 0; n < 8; ++n) acc[n] = splat8(biasD[n]);
#pragma unroll
        for (int n = 0; n < 8; ++n) acc[n] = wmma4(Ax, BihD[n], acc[n]);
#pragma unroll
        for (int c = 0; c < 32; ++c) {
#pragma unroll
            for (int n = 0; n < 8; ++n) {
                const v2f Bv = *(const v2f*)&s.W[n * 16 + l16][4 * c + koff];
                acc[n] = wmma4(hA[c], Bv, acc[n]);
            }
        }
#pragma unroll
        for (int n = 0; n < 8; ++n) {
#pragma unroll
            for (int v = 0; v < 8; ++v) {
                const int row = hi ? v + 8 : v;
                s.stage[wave][row][n * 16 + l16] = fast_tanh(acc[n][v]);
            }
        }
#pragma unroll
        for (int c = 0; c < 32; ++c)
            hA[c] = *(const v2f*)&s.stage[wave][l16][4 * c + koff];

        // pred = h_new @ fc_W^T + fc_b  (single N-tile, cols 0..2 valid)
        v8f pr = splat8(fcb);
#pragma unroll
        for (int c = 0; c < 32; ++c) {
            const v2f Bv = mk2(s.fcT[4 * c + koff][l16], s.fcT[4 * c + koff + 1][l16]);
            pr = wmma4(hA[c], Bv, pr);
        }

        // store predictions (C layout: lane o = l16, rows v / v+8) and stage them
        // as the next autoregressive input (reference carries pred as next inp).
        if (l16 < O_SIZE) {
#pragma unroll
            for (int v = 0; v < 8; ++v) {
                const int row = hi ? v + 8 : v;
                const int n  = r0 + row;
                const int bb = n >> 6, aa = n & 63;
                out[((size_t)bb * PRED + p) * (AGENTS * O_SIZE) + aa * O_SIZE + l16] = pr[v];
                s.stage[wave][row][l16] = pr[v];
            }
        }
        Ax = hi ? mk2(s.stage[wave][l16][2], 0.0f)
                : mk2(s.stage[wave][l16][0], s.stage[wave][l16][1]);
    }
}

extern "C" void kernel_launch(void* const* d_in, const int* in_sizes, int n_in,
                              void* d_out, int out_size, void* d_ws, size_t ws_size,
                              hipStream_t stream) {
    const float* src      = (const float*)d_in[0];
    const float* W_ih_enc = (const float*)d_in[1];
    const float* W_hh_enc = (const float*)d_in[2];
    const float* b_ih_enc = (const float*)d_in[3];
    const float* b_hh_enc = (const float*)d_in[4];
    const float* W_ih_dec = (const float*)d_in[5];
    const float* W_hh_dec = (const float*)d_in[6];
    const float* b_ih_dec = (const float*)d_in[7];
    const float* b_hh_dec = (const float*)d_in[8];
    const float* fc_W     = (const float*)d_in[9];
    const float* fc_b     = (const float*)d_in[10];
    // d_in[11] = pred_len (== 12, hardcoded as PRED)
    float* out = (float*)d_out;

    traj_rnn_kernel<<<NBLOCKS, 256, 0, stream>>>(
        src, W_ih_enc, W_hh_enc, b_ih_enc, b_hh_enc,
        W_ih_dec, W_hh_dec, b_ih_dec, b_hh_dec, fc_W, fc_b, out);
}